// RadixAttention_47278999995011
// MI455X (gfx1250) — compile-verified
//
#include <hip/hip_runtime.h>
#include <math.h>

// ---------------------------------------------------------------------------
// GQA decode attention (radix/paged decode step) for MI455X / gfx1250.
// B=32, S=4096, HQ=32, HKV=8, D=128, G=4. fp32 in/out.
// Memory-bound: ~1.07GB KV traffic -> ~46us at 23.3 TB/s. Strategy:
//   - TDM (tensor_load_to_lds) streams 16x128 K/V tiles into LDS, 2-deep
//     pipelined per wave with s_wait_tensorcnt.
//   - V_WMMA_F32_16X16X4_F32 for QK^T and PV (keeps VALU free for softmax).
//   - flash-decode online softmax with tanh logit capping.
// ---------------------------------------------------------------------------

typedef float        v2f   __attribute__((ext_vector_type(2)));
typedef float        v8f   __attribute__((ext_vector_type(8)));
typedef float        f4    __attribute__((ext_vector_type(4)));
typedef unsigned int u32x4 __attribute__((ext_vector_type(4)));
typedef int          i32x8 __attribute__((ext_vector_type(8)));
typedef int          i32x4 __attribute__((ext_vector_type(4)));

#define B_    32
#define S_    4096
#define HQ_   32
#define HKV_  8
#define D_    128
#define G_    4
#define NW    4                         // waves per block (wave32)
#define SCALE_ 0.08838834764831845f     // 1/sqrt(128)
#define CAP_   30.0f
#define NEG_   (-1.0e30f)

// LDS layout (float offsets)
#define QS_OFF    0                     // 16 x 128 padded Q tile
#define P_OFF     2048                  // NW x (16x16) probability tiles
#define KV_OFF    3072                  // NW x [k0,k1,v0,v1] x 2048 floats
#define RED_OFF   35840                 // NW x G x D partial outputs
#define ML_OFF    37888                 // NW x G x {m,l}
#define SNEW_OFF  37920                 // G new-token logits
#define SMEM_FLOATS 37928

#if defined(__has_builtin)
#if __has_builtin(__builtin_amdgcn_tensor_load_to_lds) && __has_builtin(__builtin_amdgcn_s_wait_tensorcnt)
#define USE_TDM 1
#endif
#endif
#ifndef USE_TDM
#define USE_TDM 0
#endif

#if USE_TDM
// Issue one 16x128 f32 tile load (row stride HKV_*D_ elements) into LDS.
// 6-arg builtin form: (u32x4 g0, i32x8 g1, i32x4 g2, i32x4 g3, i32x8 extra, i32 cpol)
__device__ __forceinline__ void tdm_issue_tile(const float* gp, unsigned lds_byte_off) {
  unsigned long long ga = (unsigned long long)(const void*)gp;
  u32x4 g0;
  g0[0] = 1u;                                   // count=1 (valid), user mode
  g0[1] = lds_byte_off;                         // lds_addr (bytes)
  g0[2] = (unsigned)(ga & 0xFFFFFFFFull);       // global_addr[31:0]
  g0[3] = (unsigned)((ga >> 32) & 0x01FFFFFFull) | (2u << 30); // [56:32] | type=2
  i32x8 g1;
  g1[0] = 0x00020000;                           // data_size=2 (4 bytes)
  g1[1] = (int)(128u << 16);                    // tensor_dim0 = 128 (low 16 bits)
  g1[2] = (int)(4096u << 16);                   // tensor_dim1 = 4096 (low 16 bits)
  g1[3] = (int)(128u << 16);                    // tile_dim0 = 128
  g1[4] = 16;                                   // tile_dim1 = 16, tile_dim2 = 0
  g1[5] = (int)(HKV_ * D_);                     // tensor_dim0_stride = 1024 elems
  g1[6] = 0;
  g1[7] = 0;
  i32x4 g2 = {0, 0, 0, 0};
  i32x4 g3 = {0, 0, 0, 0};
  i32x8 gx = {0, 0, 0, 0, 0, 0, 0, 0};
  __builtin_amdgcn_tensor_load_to_lds(g0, g1, g2, g3, gx, 0);
}
#else
// Fallback: wave-cooperative global->VGPR->LDS staging (b128 loads/stores).
__device__ __forceinline__ void wave_load_tile(const float* __restrict__ gp,
                                               float* lds, int lane) {
#pragma unroll
  for (int r = 0; r < 16; ++r) {
    f4 val = *(const f4*)(gp + (size_t)r * (HKV_ * D_) + lane * 4);
    *(f4*)(lds + r * D_ + lane * 4) = val;
  }
}
#endif

__global__ __launch_bounds__(NW * 32)
void radix_attn_decode(const float* __restrict__ q,  const float* __restrict__ kn,
                       const float* __restrict__ vn, const float* __restrict__ kcache,
                       const float* __restrict__ vcache, const int* __restrict__ seq_lens,
                       float* __restrict__ out) {
  extern __shared__ float smem[];
  const int tid  = threadIdx.x;
  const int wave = tid >> 5;
  const int lane = tid & 31;
  const int row  = lane & 15;   // WMMA M-row (A) / N-col (B,C)
  const int hi   = lane >> 4;   // which K-half of the lane layout
  const int b = blockIdx.x / HKV_;
  const int h = blockIdx.x % HKV_;

  const int seq_len = seq_lens[b];
  const int n_cache = seq_len - 1;             // cache positions (new token handled separately)

  // Stage Q (zero-padded to 16 rows) and zero P staging tiles (rows 4..15 stay 0).
  for (int i = tid; i < 16 * D_; i += NW * 32) {
    int r = i >> 7, d = i & 127;
    smem[QS_OFF + i] = (r < G_) ? q[((size_t)b * HQ_ + (h * G_ + r)) * D_ + d] : 0.0f;
  }
  for (int i = tid; i < NW * 256; i += NW * 32) smem[P_OFF + i] = 0.0f;
  __syncthreads();

  const float* qs = smem + QS_OFF;
  float* pw = smem + P_OFF + wave * 256;                 // this wave's 16x16 P tile
  const unsigned kv_base_f = KV_OFF + wave * 8192u;      // k0,k1,v0,v1 (2048 floats each)
  const size_t bh_elem = (size_t)b * S_ * HKV_ * D_ + (size_t)h * D_;

  float m[G_], l[G_];
  v8f acc[8];
  {
    v8f z = {};
#pragma unroll
    for (int j = 0; j < 8; ++j) acc[j] = z;
  }
#pragma unroll
  for (int g = 0; g < G_; ++g) { m[g] = NEG_; l[g] = 0.0f; }

  // ---- per-tile compute: QK^T (WMMA) -> online softmax -> PV (WMMA) ----
  auto process_tile = [&](int t, int buf) {
    const float* ks = smem + kv_base_f + (unsigned)buf * 2048u;
    const float* vs = smem + kv_base_f + 4096u + (unsigned)buf * 2048u;

    // scores[16x16]: rows = padded queries, cols = 16 keys.
    v8f c = {};
#pragma unroll
    for (int kc = 0; kc < 32; ++kc) {
      v2f a  = *(const v2f*)(qs + row * 128 + 4 * kc + 2 * hi);   // A: Q[row][d..d+1]
      v2f bb = *(const v2f*)(ks + row * 128 + 4 * kc + 2 * hi);   // B: K[col][d..d+1]
      c = __builtin_amdgcn_wmma_f32_16x16x4_f32(false, a, false, bb,
                                                (short)0, c, false, false);
    }

    const int  key   = t * 16 + row;
    const bool valid = (hi == 0) && (key < n_cache);
    float alpha[G_];
#pragma unroll
    for (int g = 0; g < G_; ++g) {
      float s = c[g] * SCALE_;
      s = CAP_ * tanhf(s * (1.0f / CAP_));
      s = valid ? s : NEG_;
      float rm = s;
#pragma unroll
      for (int off = 16; off >= 1; off >>= 1) rm = fmaxf(rm, __shfl_xor(rm, off, 32));
      float mn = fmaxf(m[g], rm);
      float pe = valid ? __expf(s - mn) : 0.0f;
      float rs = pe;
#pragma unroll
      for (int off = 16; off >= 1; off >>= 1) rs += __shfl_xor(rs, off, 32);
      alpha[g] = __expf(m[g] - mn);
      l[g] = l[g] * alpha[g] + rs;
      m[g] = mn;
      if (hi == 0) pw[g * 16 + row] = pe;   // A-layout staging for PV
    }

    // rescale accumulators (rows 4..15 of D are zero; only g rows matter)
#pragma unroll
    for (int j = 0; j < 8; ++j)
#pragma unroll
      for (int g = 0; g < G_; ++g) acc[j][g] *= alpha[g];

    // PV: out[g][16j+n] += sum_k P[g][k] * V[k][16j+n]
#pragma unroll
    for (int j = 0; j < 8; ++j) {
      v8f cj = acc[j];
#pragma unroll
      for (int kc = 0; kc < 4; ++kc) {
        v2f a = *(const v2f*)(pw + row * 16 + 4 * kc + 2 * hi);
        int r = 4 * kc + 2 * hi;
        v2f bb;
        bb.x = vs[r * 128 + 16 * j + row];
        bb.y = vs[(r + 1) * 128 + 16 * j + row];
        cj = __builtin_amdgcn_wmma_f32_16x16x4_f32(false, a, false, bb,
                                                   (short)0, cj, false, false);
      }
      acc[j] = cj;
    }
  };

  const int n_tiles = (n_cache + 15) >> 4;
  const int my_nt = (n_tiles > wave) ? ((n_tiles - wave + NW - 1) / NW) : 0;

#if USE_TDM
  if (my_nt > 0) {
    tdm_issue_tile(kcache + bh_elem, (kv_base_f) * 4u);
    tdm_issue_tile(vcache + bh_elem, (kv_base_f + 4096u) * 4u);
  }
  for (int i = 0; i < my_nt; ++i) {
    const int t = wave + i * NW;
    if (i + 1 < my_nt) {
      const size_t nxt = bh_elem + (size_t)((t + NW) * 16) * (HKV_ * D_);
      const unsigned nb = (unsigned)((i + 1) & 1) * 2048u;
      tdm_issue_tile(kcache + nxt, (kv_base_f + nb) * 4u);
      tdm_issue_tile(vcache + nxt, (kv_base_f + 4096u + nb) * 4u);
      __builtin_amdgcn_s_wait_tensorcnt(2);   // current tile's 2 loads retired
    } else {
      __builtin_amdgcn_s_wait_tensorcnt(0);
    }
    process_tile(t, i & 1);
  }
#else
  for (int i = 0; i < my_nt; ++i) {
    const int t = wave + i * NW;
    const size_t base = bh_elem + (size_t)(t * 16) * (HKV_ * D_);
    wave_load_tile(kcache + base, smem + kv_base_f, lane);
    wave_load_tile(vcache + base, smem + kv_base_f + 4096u, lane);
    process_tile(t, 0);
  }
#endif

  // ---- publish per-wave partials ----
  if (hi == 0) {
#pragma unroll
    for (int j = 0; j < 8; ++j)
#pragma unroll
      for (int g = 0; g < G_; ++g)
        smem[RED_OFF + (wave * G_ + g) * D_ + 16 * j + row] = acc[j][g];
    if (lane == 0) {
#pragma unroll
      for (int g = 0; g < G_; ++g) {
        smem[ML_OFF + wave * 8 + g * 2 + 0] = m[g];
        smem[ML_OFF + wave * 8 + g * 2 + 1] = l[g];
      }
    }
  }
  __syncthreads();

  // ---- new-token logit: wave g computes dot(q_g, k_new) ----
  {
    const int g = wave;  // NW == G_
    float part = 0.0f;
#pragma unroll
    for (int ii = 0; ii < 4; ++ii) {
      int d = lane * 4 + ii;
      part += qs[g * D_ + d] * kn[((size_t)b * HKV_ + h) * D_ + d];
    }
#pragma unroll
    for (int off = 16; off >= 1; off >>= 1) part += __shfl_xor(part, off, 32);
    if (lane == 0) smem[SNEW_OFF + g] = CAP_ * tanhf(part * SCALE_ * (1.0f / CAP_));
  }
  __syncthreads();

  // ---- combine waves + new token, write output ----
  for (int idx = tid; idx < G_ * D_; idx += NW * 32) {
    const int g = idx >> 7, d = idx & 127;
    const float sn = smem[SNEW_OFF + g];
    float M = sn;
#pragma unroll
    for (int w = 0; w < NW; ++w) M = fmaxf(M, smem[ML_OFF + w * 8 + g * 2]);
    float L = 0.0f, o = 0.0f;
#pragma unroll
    for (int w = 0; w < NW; ++w) {
      float e = __expf(smem[ML_OFF + w * 8 + g * 2] - M);
      L += smem[ML_OFF + w * 8 + g * 2 + 1] * e;
      o += smem[RED_OFF + (w * G_ + g) * D_ + d] * e;
    }
    const float en = __expf(sn - M);
    L += en;
    o += en * vn[((size_t)b * HKV_ + h) * D_ + d];
    out[((size_t)b * HQ_ + (h * G_ + g)) * D_ + d] = o / L;
  }
}

extern "C" void kernel_launch(void* const* d_in, const int* in_sizes, int n_in,
                              void* d_out, int out_size, void* d_ws, size_t ws_size,
                              hipStream_t stream) {
  (void)in_sizes; (void)n_in; (void)out_size; (void)d_ws; (void)ws_size;
  const float* q  = (const float*)d_in[0];
  const float* k  = (const float*)d_in[1];
  const float* v  = (const float*)d_in[2];
  const float* kc = (const float*)d_in[3];
  const float* vc = (const float*)d_in[4];
  const int*   sl = (const int*)d_in[5];
  float* out = (float*)d_out;

  dim3 grid(B_ * HKV_);
  dim3 block(NW * 32);
  size_t smem_bytes = (size_t)SMEM_FLOATS * sizeof(float);
  radix_attn_decode<<<grid, block, smem_bytes, stream>>>(q, k, v, kc, vc, sl, out);
}